// ShapeNet1D_L2_Global_49632642072830
// MI455X (gfx1250) — compile-verified
//
#include <hip/hip_runtime.h>
#include <stdint.h>

// ---------------------------------------------------------------------------
// ShapeNet1D hat-basis evaluation, MI455X (gfx1250).
// out[i,j] = l2(j-1) - l2(j),  l2(e) = relu(1 - relu(xf[e+1]-x)*inv[e]),
// with sentinels l2(-1)=1, l2(2048)=0.  Pure streaming-write problem:
// 268.6 MB out / 23.3 TB/s ~ 11.5 us.  No matmul structure -> no WMMA;
// use gfx1250 async global->LDS staging + NT coalesced b32 stores instead.
// ---------------------------------------------------------------------------

#ifndef __has_builtin
#define __has_builtin(x) 0
#endif

#if __has_builtin(__builtin_amdgcn_global_load_async_to_lds_b128)
#define USE_ASYNC_LDS 1
#else
#define USE_ASYNC_LDS 0
#endif

#define NPTS   32768
#define NELEM  2048
#define NNODES 2049
#define TPB    256
#define ROWS_PER_BLOCK 32

// Builtin signature (from hipcc diagnostic): (int4 AS1*, int4 AS3*, imm, imm)
typedef int v4i __attribute__((vector_size(16)));
typedef __attribute__((address_space(1))) v4i as1_v4i;
typedef __attribute__((address_space(3))) v4i as3_v4i;

__device__ __forceinline__ void stage_b128(const float* g, float* l) {
#if USE_ASYNC_LDS
  // gfx1250 async DMA: global -> LDS, 16B per lane, tracked by ASYNCcnt.
  // Generic->AS casts done via integer round-trip (AS3 ptr = low 32 bits).
  __builtin_amdgcn_global_load_async_to_lds_b128(
      (as1_v4i*)(uintptr_t)g, (as3_v4i*)(uint32_t)(uintptr_t)l, 0, 0);
#else
  float4 v = *(const float4*)g;
  *(float4*)l = v;
#endif
}

__device__ __forceinline__ void wait_async_all() {
#if USE_ASYNC_LDS
#if __has_builtin(__builtin_amdgcn_s_wait_asynccnt)
  __builtin_amdgcn_s_wait_asynccnt(0);
#else
  asm volatile("s_wait_asynccnt 0x0" ::: "memory");
#endif
#endif
}

__global__ __launch_bounds__(TPB) void hat_basis_kernel(
    const float* __restrict__ x_eval,   // [NPTS]
    const float* __restrict__ x_full,   // [NNODES]
    float* __restrict__ out)            // [NPTS, NNODES]
{
  __shared__ __align__(16) float s_xf[NNODES + 3]; // nodes (padded)
  __shared__ float s_inv[NELEM];                   // 1/(h+1e-9)

  const int t = threadIdx.x;

  // ---- stage x_full[0..2047] via async b128 (2 chunks / thread) ----
  stage_b128(x_full + 4 * t,           s_xf + 4 * t);
  stage_b128(x_full + 4 * (t + TPB),   s_xf + 4 * (t + TPB));
  if (t == 0) s_xf[NELEM] = x_full[NELEM];  // last node, plain path
  wait_async_all();
  __syncthreads();

  // ---- per-element reciprocal (once per block; div -> fma in hot loop) ----
  for (int e = t; e < NELEM; e += TPB)
    s_inv[e] = 1.0f / ((s_xf[e + 1] - s_xf[e]) + 1e-9f);
  __syncthreads();

  // ---- hoist this thread's column constants into registers ----
  // thread t owns columns j = t + 256*s  (coalesced b32 stores per wave)
  float A[8], Ai[8], B[8], Bi[8];
#pragma unroll
  for (int s = 0; s < 8; ++s) {
    const int j = t + TPB * s;
    // j==0 sentinel: A=-1 => relu(A-x)=0 => l2m = 1 (x_eval >= 0)
    A[s]  = (j == 0) ? -1.0f : s_xf[j];        // xb of element j-1
    Ai[s] = (j == 0) ?  0.0f : s_inv[j - 1];
    B[s]  = s_xf[j + 1];                        // xb of element j
    Bi[s] = s_inv[j];
  }
  const bool lastCol = (t == 0);                // column 2048 = l2(2047)
  float XL = 0.0f, IL = 0.0f;
  if (lastCol) { XL = s_xf[NELEM]; IL = s_inv[NELEM - 1]; }

  // ---- row loop: 1 scalar K$ load + 8 FMA-chains + 8 NT stores per thread ----
  const int row0 = blockIdx.x * ROWS_PER_BLOCK;
  for (int r = 0; r < ROWS_PER_BLOCK; ++r) {
    const int row = row0 + r;
    const float x = x_eval[row];                        // uniform -> s_load
    float* orow = out + (size_t)row * NNODES;
#pragma unroll
    for (int s = 0; s < 8; ++s) {
      const int j = t + TPB * s;
      const float l2m = fmaxf(1.0f - fmaxf(A[s] - x, 0.0f) * Ai[s], 0.0f);
      const float l2c = fmaxf(1.0f - fmaxf(B[s] - x, 0.0f) * Bi[s], 0.0f);
      __builtin_nontemporal_store(l2m - l2c, orow + j);
    }
    if (lastCol) {
      const float l2 = fmaxf(1.0f - fmaxf(XL - x, 0.0f) * IL, 0.0f);
      __builtin_nontemporal_store(l2, orow + NELEM);
    }
  }
}

extern "C" void kernel_launch(void* const* d_in, const int* in_sizes, int n_in,
                              void* d_out, int out_size, void* d_ws, size_t ws_size,
                              hipStream_t stream) {
  (void)n_in; (void)out_size; (void)d_ws; (void)ws_size;
  const float* x_eval = (const float*)d_in[0];   // 32768 floats
  const float* x_full = (const float*)d_in[1];   // 2049 floats
  float* out = (float*)d_out;                    // 32768 x 2049 floats

  const int n_points = in_sizes[0];              // 32768
  const int n_blocks = n_points / ROWS_PER_BLOCK;

  hat_basis_kernel<<<dim3(n_blocks), dim3(TPB), 0, stream>>>(x_eval, x_full, out);
}